// ALiBi_34170759807523
// MI455X (gfx1250) — compile-verified
//
#include <hip/hip_runtime.h>
#include <stdint.h>

// ALiBi bias subtraction: out[b,h,i,j] = scores[b,h,i,j] - slopes[h] * (pos[i] - pos[j])
// (the scalar `offset` cancels in pos_i - pos_j, so it is not needed numerically).
//
// Shapes (reference, fixed): scores [B=4, H=16, S=2048, S=2048] fp32,
// slopes [16], positions [4096]. Memory-bound: 2 GiB traffic -> ~92us floor
// at 23.3 TB/s HBM. Strategy:
//   * 16 consecutive rows (same head) per 256-thread block -> 8192 blocks,
//     each streaming 128 KB of scores; slope is block-uniform.
//   * CDNA5 async global->LDS copy (global_load_async_to_lds_b128 +
//     s_wait_asynccnt) stages positions[0:S] (8 KB) once per block; p_j chunks
//     are then hoisted into registers (identical across the 16 rows).
//   * b128 non-temporal load/store on the streamed scores/out so the one-pass
//     2 GiB stream does not evict the hot positions/slopes lines from L2.
//   * S, H, rows-per-block are constexpr powers of two -> index math is
//     mask/shift, no SALU division sequence.

typedef float v4f __attribute__((ext_vector_type(4)));

constexpr int TPB  = 256;   // threads per block (8 wave32 waves)
constexpr int S    = 2048;  // seq_len = scores.shape[2]
constexpr int H    = 16;    // num heads
constexpr int ROWS = 16;    // rows per block (16 | 2048 -> never crosses a head)
constexpr int VPT  = S / (4 * TPB);  // v4f chunks per thread per row = 2

__global__ __launch_bounds__(TPB) void alibi_kernel(const float* __restrict__ scores,
                                                    const float* __restrict__ slopes,
                                                    const float* __restrict__ positions,
                                                    float* __restrict__ out) {
    __shared__ float spos[S];                    // staged positions[0:S], 8 KB

    const int t  = threadIdx.x;
    const long long r0 = (long long)blockIdx.x * ROWS;  // first row of this block
    const int i0 = (int)(r0 & (S - 1));                 // r0 % S   (mask)
    const int h  = (int)((r0 >> 11) & (H - 1));         // (r0/S)%H (shift+mask)

    // ---- CDNA5 async global->LDS staging of positions[0:S] ------------------
    // Each of the 256 lanes copies two 16B chunks (2 * 256 * 16B = 8 KB).
    {
        // Low 32 bits of a generic pointer into LDS == LDS byte address
        // (ISA 10.2, LDS aperture).
        unsigned lds0 = (unsigned)(uintptr_t)spos + (unsigned)t * 16u;
        unsigned lds1 = lds0 + TPB * 16u;
        const float* g0 = positions + t * 4;
        const float* g1 = g0 + TPB * 4;
        asm volatile("global_load_async_to_lds_b128 %0, %1, off"
                     :: "v"(lds0), "v"(g0) : "memory");
        asm volatile("global_load_async_to_lds_b128 %0, %1, off"
                     :: "v"(lds1), "v"(g1) : "memory");
        asm volatile("s_wait_asynccnt 0" ::: "memory");
    }
    __syncthreads();                             // make all waves' LDS writes visible

    const float slope = slopes[h];               // block-uniform -> scalar load

    // p_j chunks are the same for every row this block touches: hoist out of
    // the row loop into registers (2 x v4f = 8 VGPRs, 2 ds_load_b128 total).
    const v4f* __restrict__ prow = (const v4f*)spos;
    v4f pj[VPT];
#pragma unroll
    for (int k = 0; k < VPT; ++k) pj[k] = prow[t + k * TPB];

#pragma unroll
    for (int row = 0; row < ROWS; ++row) {
        const float pi = spos[i0 + row];         // query position for this row
        const size_t off = (size_t)(r0 + row) * (size_t)S;
        const v4f* __restrict__ srow = (const v4f*)(scores + off);
        v4f*       __restrict__ orow = (v4f*)(out + off);
#pragma unroll
        for (int k = 0; k < VPT; ++k) {
            const int idx = t + k * TPB;
            v4f s = __builtin_nontemporal_load(&srow[idx]);  // th:NT streamed read
            v4f o = s - slope * (pi - pj[k]);                // elementwise FMA
            __builtin_nontemporal_store(o, &orow[idx]);      // th:NT streamed write
        }
    }
}

extern "C" void kernel_launch(void* const* d_in, const int* in_sizes, int n_in,
                              void* d_out, int out_size, void* d_ws, size_t ws_size,
                              hipStream_t stream) {
    const float* scores    = (const float*)d_in[0];
    const float* slopes    = (const float*)d_in[1];
    const float* positions = (const float*)d_in[2];
    // d_in[3] = offset: cancels in (pos_i+off)-(pos_j+off); unused.
    float* out = (float*)d_out;

    const long long rows   = (long long)out_size / S;   // B*H*S = 131072
    const long long blocks = rows / ROWS;               // 8192 blocks

    dim3 grid((unsigned)blocks), block(TPB);
    hipLaunchKernelGGL(alibi_kernel, grid, block, 0, stream,
                       scores, slopes, positions, out);
}